// SeedCondExpansion_85839216377873
// MI455X (gfx1250) — compile-verified
//
#include <hip/hip_runtime.h>
#include <math.h>
#include <stdint.h>

#define BB 4
#define NN 4096
#define DD 256
#define PP 64
#define ALPHA 0.2f
#define KCAP 255

typedef float v2f __attribute__((ext_vector_type(2)));
typedef float v8f __attribute__((ext_vector_type(8)));

// async-to-LDS builtin takes pointers to 4 x i32 vectors in AS(1)/AS(3)
typedef int v4i __attribute__((vector_size(16)));
typedef __attribute__((address_space(1))) v4i gv4i;
typedef __attribute__((address_space(3))) v4i lv4i;

#if __has_builtin(__builtin_amdgcn_global_load_async_to_lds_b128)
#define HAVE_ASYNC_LDS 1
#endif

// ---------------- Kernel 1: seed argmax, q = x[b,seed]@Wq, theta MLP ----------------
__global__ void prep_kernel(const float* __restrict__ x, const float* __restrict__ cm,
                            const float* __restrict__ Wq,
                            const float* __restrict__ sctx, const float* __restrict__ lst,
                            const float* __restrict__ w1, const float* __restrict__ b1,
                            const float* __restrict__ w2, const float* __restrict__ b2,
                            float* __restrict__ q_ws, float* __restrict__ th_ws) {
  __shared__ float sv[256];
  __shared__ int   si[256];
  const int b = blockIdx.x, t = threadIdx.x;

  // argmax(cluster_mask) -> first index of max
  float bv = -1.0f; int bi = 0;
  for (int n = t; n < NN; n += 256) {
    float v = cm[b * NN + n];
    if (v > bv) { bv = v; bi = n; }
  }
  sv[t] = bv; si[t] = bi;
  __syncthreads();
  for (int s = 128; s > 0; s >>= 1) {
    if (t < s) {
      float v2 = sv[t + s]; int i2 = si[t + s];
      if (v2 > sv[t] || (v2 == sv[t] && i2 < si[t])) { sv[t] = v2; si[t] = i2; }
    }
    __syncthreads();
  }
  const int seed = si[0];
  __syncthreads();

  // q[b,:] = x[b,seed,:] @ Wq
  if (t < PP) {
    const float* xr = x + ((size_t)b * NN + seed) * DD;
    float acc = 0.f;
    for (int d = 0; d < DD; ++d) acc += xr[d] * Wq[d * PP + t];
    q_ws[b * PP + t] = acc;
  }

  // theta = relu(concat(sctx,lst) @ w1 + b1) @ w2 + b2
  float part = 0.f;
  if (t < 128) {
    float acc = b1[t];
    for (int i = 0; i < 512; ++i) {
      float in = (i < 384) ? sctx[b * 384 + i] : lst[b * 128 + (i - 384)];
      acc += in * w1[i * 128 + t];
    }
    acc = fmaxf(acc, 0.f);
    part = acc * w2[t];
  }
  sv[t] = part;
  __syncthreads();
  for (int s = 128; s > 0; s >>= 1) { if (t < s) sv[t] += sv[t + s]; __syncthreads(); }
  if (t == 0) th_ws[b] = sv[0] + b2[0];
}

// ---------------- Kernel 2: fused e2c = adj@cm and deg = max(rowsum,1) ----------------
// One block per (b,row): a single streaming pass over the 256MB adjacency (the
// bandwidth roofline term, ~11us at 23.3 TB/s). Coalesced float4 loads + prefetch.
__global__ void adj_kernel(const float* __restrict__ adj, const float* __restrict__ cm,
                           float* __restrict__ e2c, float* __restrict__ deg) {
  __shared__ float s0s[256], s1s[256];
  const int t = threadIdx.x;
  const int b = blockIdx.x >> 12;
  const int n = blockIdx.x & (NN - 1);
  const float4* __restrict__ row = (const float4*)(adj + ((size_t)b * NN + n) * NN);
  const float4* __restrict__ c4  = (const float4*)(cm + b * NN);
  float s0 = 0.f, s1 = 0.f;
  for (int i = t; i < NN / 4; i += 256) {
    if (i + 256 < NN / 4) __builtin_prefetch((const void*)(row + i + 256), 0, 0);
    float4 a = row[i]; float4 c = c4[i];
    s0 += a.x + a.y + a.z + a.w;
    s1 += a.x * c.x + a.y * c.y + a.z * c.z + a.w * c.w;
  }
  s0s[t] = s0; s1s[t] = s1;
  __syncthreads();
  for (int s = 128; s > 0; s >>= 1) {
    if (t < s) { s0s[t] += s0s[t + s]; s1s[t] += s1s[t + s]; }
    __syncthreads();
  }
  if (t == 0) { e2c[b * NN + n] = s1s[0]; deg[b * NN + n] = fmaxf(s0s[0], 1.0f); }
}

// ---------------- Kernel 3: scores via WMMA f32 16x16x4, Wk staged via async-to-LDS ----
// One wave per 16-row tile of x; 4 waves per block share Wk (64KB) in LDS.
// A layout (ISA 7.12.2, 32-bit A 16x4): lanes 0-15 hold K=0,1 ; lanes 16-31 hold K=2,3.
__global__ void scores_wmma_kernel(const float* __restrict__ x, const float* __restrict__ Wk,
                                   const float* __restrict__ q_ws, float* __restrict__ scores) {
  __shared__ float lds_wk[DD * PP];   // 64KB: Wk row-major
  const int tid  = threadIdx.x;       // 0..127
  const int wave = tid >> 5;
  const int lane = tid & 31;
  const int b    = blockIdx.x >> 6;   // 64 blocks per batch (block = 4 waves = 64 rows)
  const int tile = blockIdx.x & 63;
  const int n0   = tile * 64 + wave * 16;
  const int r    = lane & 15;
  const int half = lane >> 4;

  // ---- Stage Wk into LDS (64KB), preferring the CDNA5 async-copy path ----
#ifdef HAVE_ASYNC_LDS
#pragma unroll
  for (int i = 0; i < (DD * PP) / (4 * 128); ++i) {   // 32 iters of b128 per lane
    const int f = tid + i * 128;                      // float4 index
    gv4i* gp = (gv4i*)(unsigned long long)(uintptr_t)(Wk + 4 * f);
    lv4i* lp = (lv4i*)(unsigned int)(uintptr_t)(&lds_wk[4 * f]);
    __builtin_amdgcn_global_load_async_to_lds_b128(gp, lp, 0, 0);
  }
#if __has_builtin(__builtin_amdgcn_s_wait_asynccnt)
  __builtin_amdgcn_s_wait_asynccnt(0);
#else
  asm volatile("s_wait_asynccnt 0" ::: "memory");
#endif
#else
  for (int i = 0; i < (DD * PP) / (4 * 128); ++i) {
    const int f = tid + i * 128;
    *(float4*)(&lds_wk[4 * f]) = *(const float4*)(Wk + 4 * f);
  }
#endif
  __syncthreads();

  const float* __restrict__ xrow = x + ((size_t)b * NN + n0 + r) * DD;

  const v8f vzero = {0.f, 0.f, 0.f, 0.f, 0.f, 0.f, 0.f, 0.f};
  v8f acc[4];
  acc[0] = vzero; acc[1] = vzero; acc[2] = vzero; acc[3] = vzero;

  for (int k0 = 0; k0 < DD; k0 += 4) {
    float4 xa = *(const float4*)(xrow + k0);
    v2f A;
    A[0] = half ? xa.z : xa.x;
    A[1] = half ? xa.w : xa.y;
    const int kr = k0 + (half << 1);
#pragma unroll
    for (int tt = 0; tt < 4; ++tt) {
      v2f Bv;
      Bv[0] = lds_wk[kr * PP + tt * 16 + r];
      Bv[1] = lds_wk[(kr + 1) * PP + tt * 16 + r];
      acc[tt] = __builtin_amdgcn_wmma_f32_16x16x4_f32(
          false, A, false, Bv, (short)0, acc[tt], false, false);
    }
  }

  // fold x_k tile against q (per-batch 64-vector), then 16-lane reduction per row
  float qv[4];
#pragma unroll
  for (int tt = 0; tt < 4; ++tt) qv[tt] = q_ws[b * PP + tt * 16 + r];
#pragma unroll
  for (int j = 0; j < 8; ++j) {
    float pj = acc[0][j] * qv[0] + acc[1][j] * qv[1] + acc[2][j] * qv[2] + acc[3][j] * qv[3];
    for (int off = 8; off > 0; off >>= 1) pj += __shfl_xor(pj, off, 32);
    if (r == 0) scores[(size_t)b * NN + n0 + half * 8 + j] = pj * 0.125f;  // 1/sqrt(64)
  }
}

// ---------------- Kernel 4: sigmoid, top-KCAP trim, outputs ----------------
__global__ void finalize_kernel(const float* __restrict__ scores, const float* __restrict__ e2c,
                                const float* __restrict__ deg, const float* __restrict__ cand,
                                const float* __restrict__ th_ws,
                                float* __restrict__ out_hard, float* __restrict__ out_p,
                                float* __restrict__ out_theta, float* __restrict__ out_es,
                                float* __restrict__ out_mean) {
  __shared__ float pbuf[NN];
  __shared__ int   ired[256];
  __shared__ float fred[256];
  const int b = blockIdx.x, t = threadIdx.x;
  const size_t base = (size_t)b * NN;
  const float th = th_ws[b];

  int cnt = 0;
  for (int n = t; n < NN; n += 256) {
    float cs = scores[base + n] + ALPHA * (e2c[base + n] / deg[base + n]);
    float pv = 1.0f / (1.0f + expf(-(cs - th)));   // TAU == 1
    pv *= cand[base + n];
    pbuf[n] = pv;
    cnt += (pv > 0.5f) ? 1 : 0;
  }
  ired[t] = cnt; __syncthreads();
  for (int s = 128; s > 0; s >>= 1) { if (t < s) ired[t] += ired[t + s]; __syncthreads(); }
  const int over = ired[0];
  __syncthreads();

  if (over > KCAP) {
    // p in [0,1): nonnegative floats -> bit pattern is order-isomorphic.
    // Find rank-(KCAP-1) value (descending): smallest u with count(bits>u) <= KCAP-1.
    unsigned lo = 0u, hi = 0x3F800000u;
    while (lo < hi) {
      unsigned mid = lo + ((hi - lo) >> 1);
      int c = 0;
      for (int n = t; n < NN; n += 256) c += (__float_as_uint(pbuf[n]) > mid) ? 1 : 0;
      ired[t] = c; __syncthreads();
      for (int s = 128; s > 0; s >>= 1) { if (t < s) ired[t] += ired[t + s]; __syncthreads(); }
      int ct = ired[0]; __syncthreads();
      if (ct > KCAP - 1) lo = mid + 1; else hi = mid;
    }
    const unsigned tb = lo;
    int cg = 0;
    for (int n = t; n < NN; n += 256) cg += (__float_as_uint(pbuf[n]) > tb) ? 1 : 0;
    ired[t] = cg; __syncthreads();
    for (int s = 128; s > 0; s >>= 1) { if (t < s) ired[t] += ired[t + s]; __syncthreads(); }
    const int quota = KCAP - ired[0];
    __syncthreads();

    // Stable (lowest-index-first) selection among ties, matching jax top_k.
    const int start = t * 16;
    int ceq = 0;
    for (int k = 0; k < 16; ++k) ceq += (__float_as_uint(pbuf[start + k]) == tb) ? 1 : 0;
    ired[t] = ceq; __syncthreads();
    for (int off = 1; off < 256; off <<= 1) {
      int v = (t >= off) ? ired[t - off] : 0;
      __syncthreads();
      ired[t] += v;
      __syncthreads();
    }
    int rank = ired[t] - ceq;  // exclusive prefix of equal-count
    for (int k = 0; k < 16; ++k) {
      const int n = start + k;
      unsigned bbits = __float_as_uint(pbuf[n]);
      bool keep;
      if (bbits > tb) keep = true;
      else if (bbits == tb) { keep = (rank < quota); rank++; }
      else keep = false;
      if (!keep) pbuf[n] = 0.0f;
    }
    __syncthreads();
  }

  float s = 0.f;
  for (int n = t; n < NN; n += 256) {
    float pv = pbuf[n];
    out_p[base + n]    = pv;
    out_hard[base + n] = (pv > 0.5f) ? 1.0f : 0.0f;  // straight-through forward value
    s += pv;
  }
  fred[t] = s; __syncthreads();
  for (int ss = 128; ss > 0; ss >>= 1) { if (t < ss) fred[t] += fred[t + ss]; __syncthreads(); }
  if (t == 0) {
    out_theta[b] = th;
    out_es[b]    = fred[0];
    out_mean[b]  = fred[0] * (1.0f / (float)NN);
  }
}

extern "C" void kernel_launch(void* const* d_in, const int* in_sizes, int n_in,
                              void* d_out, int out_size, void* d_ws, size_t ws_size,
                              hipStream_t stream) {
  const float* x    = (const float*)d_in[0];
  const float* adj  = (const float*)d_in[1];
  const float* sctx = (const float*)d_in[2];
  const float* lst  = (const float*)d_in[3];
  const float* cm   = (const float*)d_in[4];
  const float* cand = (const float*)d_in[5];
  const float* Wq   = (const float*)d_in[6];
  const float* Wk   = (const float*)d_in[7];
  const float* w1   = (const float*)d_in[8];
  const float* b1   = (const float*)d_in[9];
  const float* w2   = (const float*)d_in[10];
  const float* b2   = (const float*)d_in[11];

  float* out = (float*)d_out;
  float* ws  = (float*)d_ws;
  float* e2c    = ws;                 // B*N
  float* deg    = ws + BB * NN;       // B*N
  float* scores = ws + 2 * BB * NN;   // B*N
  float* q_ws   = ws + 3 * BB * NN;   // B*64
  float* th_ws  = q_ws + BB * PP;     // B

  float* out_hard  = out;
  float* out_p     = out + BB * NN;
  float* out_theta = out + 2 * BB * NN;
  float* out_es    = out_theta + BB;
  float* out_mean  = out_es + BB;

  prep_kernel<<<BB, 256, 0, stream>>>(x, cm, Wq, sctx, lst, w1, b1, w2, b2, q_ws, th_ws);
  adj_kernel<<<BB * NN, 256, 0, stream>>>(adj, cm, e2c, deg);
  scores_wmma_kernel<<<BB * (NN / 64), 128, 0, stream>>>(x, Wk, q_ws, scores);
  finalize_kernel<<<BB, 256, 0, stream>>>(scores, e2c, deg, cand, th_ws,
                                          out_hard, out_p, out_theta, out_es, out_mean);
}